// SpanRelationLBP_44659069943943
// MI455X (gfx1250) — compile-verified
//
#include <hip/hip_runtime.h>
#include <stdint.h>

// ---------------------------------------------------------------------------
// SpanRelationLBP on gfx1250.
// Memory-bound streaming LBP: fused per-iteration kernels read shstoh/shstot
// exactly once, recompute the 2x2 log-potentials in-register, and stage all
// streamed data through LDS with CDNA5 async-to-LDS loads in a 4-deep ring
// (3 steps in flight, s_wait_asynccnt throttled). Messages are stored with
// non-temporal hints so the 141 MB factor+mask set stays L2-resident across
// the 3 LBP iterations. Only m_fnh_h / m_fnt_t persist in workspace (~57 MB).
// ---------------------------------------------------------------------------

#define T96   96
#define B4    4
#define NP    (T96 * T96)       // 9216 positions per (b,x)-plane
#define NSTEP (NP / 256)        // 36 pipelined steps per block
#define NQ    (B4 * T96 * T96 * 2)
#define NLOG2 0.69314718055994530942f
#define FEPS  1.1920929e-07f

typedef float f2v __attribute__((ext_vector_type(2)));

__device__ __forceinline__ unsigned lds_off(const void* p) {
  // generic LDS address = {aperture, offset}; low 32 bits are the LDS offset
  return (unsigned)(uintptr_t)p;
}

// CDNA5 async global->LDS (VGLOBAL GVS mode: saddr base + 32-bit vgpr offset)
__device__ __forceinline__ void async_b128(unsigned lds, unsigned voff, const void* sbase) {
  asm volatile("global_load_async_to_lds_b128 %0, %1, %2"
               :: "v"(lds), "v"(voff), "s"(sbase) : "memory");
}
__device__ __forceinline__ void async_b64(unsigned lds, unsigned voff, const void* sbase) {
  asm volatile("global_load_async_to_lds_b64 %0, %1, %2"
               :: "v"(lds), "v"(voff), "s"(sbase) : "memory");
}
__device__ __forceinline__ void async_b32(unsigned lds, unsigned voff, const void* sbase) {
  asm volatile("global_load_async_to_lds_b32 %0, %1, %2"
               :: "v"(lds), "v"(voff), "s"(sbase) : "memory");
}
__device__ __forceinline__ void wait_async_9() { asm volatile("s_wait_asynccnt 0x9" ::: "memory"); }
__device__ __forceinline__ void wait_async_6() { asm volatile("s_wait_asynccnt 0x6" ::: "memory"); }
__device__ __forceinline__ void wait_async_3() { asm volatile("s_wait_asynccnt 0x3" ::: "memory"); }
__device__ __forceinline__ void wait_async_0() { asm volatile("s_wait_asynccnt 0x0" ::: "memory"); }

__device__ __forceinline__ float lse2(float a, float b) {
  float mx = fmaxf(a, b), mn = fminf(a, b);
  return mx + __logf(1.0f + __expf(mn - mx));
}

// softmax over 4 joint states + eps, viewed as 2x2, edge-normalized potential
__device__ __forceinline__ void edge_f(float4 x, float& f00, float& f01, float& f10, float& f11) {
  float mx = fmaxf(fmaxf(x.x, x.y), fmaxf(x.z, x.w));
  float e0 = __expf(x.x - mx), e1 = __expf(x.y - mx);
  float e2 = __expf(x.z - mx), e3 = __expf(x.w - mx);
  float inv = __builtin_amdgcn_rcpf(e0 + e1 + e2 + e3);
  float p0 = e0 * inv + FEPS, p1 = e1 * inv + FEPS;
  float p2 = e2 * inv + FEPS, p3 = e3 * inv + FEPS;
  float lr0 = __logf(p0 + p1 + FEPS), lr1 = __logf(p2 + p3 + FEPS);
  float lc0 = __logf(p0 + p2 + FEPS), lc1 = __logf(p1 + p3 + FEPS);
  f00 = __logf(p0) - 0.5f * (lr0 + lc0);
  f01 = __logf(p1) - 0.5f * (lr0 + lc1);
  f10 = __logf(p2) - 0.5f * (lr1 + lc0);
  f11 = __logf(p3) - 0.5f * (lr1 + lc1);
}

// Fused factor update: given f, incoming (a = q_side - m_side_old) and node
// belief (b = q0), produce the new node-message (mn) and side-message (h).
__device__ __forceinline__ void bp_update(
    float f00, float f01, float f10, float f11,
    float a0, float a1, float b0, float b1,
    float& mn0, float& mn1, float& h0, float& h1) {
  float u0 = lse2(a0 + f00, a1 + f01);
  float u1 = lse2(a0 + f10, a1 + f11);
  float ln = lse2(u0, u1);
  mn0 = u0 - ln; mn1 = u1 - ln;
  float n0 = b0 - mn0, n1 = b1 - mn1;
  float v0 = lse2(n0 + f00, n1 + f10);
  float v1 = lse2(n0 + f01, n1 + f11);
  float lh = lse2(v0, v1);
  h0 = v0 - lh; h1 = v1 - lh;
}

// ---------------------------------------------------------------------------
// Initial beliefs from masks: messages start at log(1/2), so q's are -log2 *
// masked counts (no unary terms on the pre-iteration beliefs).
// ---------------------------------------------------------------------------
__global__ void __launch_bounds__(256) k_init(
    const int* __restrict__ mh, const int* __restrict__ mt,
    float* __restrict__ q0, float* __restrict__ q1, float* __restrict__ q2) {
  const int n = B4 * T96 * T96;
  int gid = blockIdx.x * 256 + threadIdx.x;
  if (gid < n) {                       // q0[b,i,j] = -log2 * (sum_k mh + sum_k mt)
    int b = gid / (T96 * T96), r = gid % (T96 * T96);
    const int* ph = mh + ((size_t)(b * T96 + r / T96) * T96 + (r % T96)) * T96;
    const int* pt = mt + ((size_t)(b * T96 + r / T96) * T96 + (r % T96)) * T96;
    int c = 0;
    for (int k = 0; k < T96; ++k) c += ph[k] + pt[k];
    float v = -NLOG2 * (float)c;
    q0[gid * 2 + 0] = v; q0[gid * 2 + 1] = v;
  } else if (gid < 2 * n) {            // q1[b,i,k] = -log2 * sum_j mh
    int g = gid - n;
    int b = g / (T96 * T96), r = g % (T96 * T96), i = r / T96, k = r % T96;
    const int* ph = mh + (size_t)(b * T96 + i) * T96 * T96 + k;
    int c = 0;
    for (int j = 0; j < T96; ++j) c += ph[(size_t)j * T96];
    float v = -NLOG2 * (float)c;
    q1[g * 2 + 0] = v; q1[g * 2 + 1] = v;
  } else if (gid < 3 * n) {            // q2[b,j,k] = -log2 * sum_i mt
    int g = gid - 2 * n;
    int b = g / (T96 * T96), r = g % (T96 * T96), j = r / T96, k = r % T96;
    const int* pt = mt + (size_t)b * T96 * T96 * T96 + (size_t)j * T96 + k;
    int c = 0;
    for (int i2 = 0; i2 < T96; ++i2) c += pt[(size_t)i2 * T96 * T96];
    float v = -NLOG2 * (float)c;
    q2[g * 2 + 0] = v; q2[g * 2 + 1] = v;
  }
}

// ---------------------------------------------------------------------------
// fnh chain: one block per (b,d1) plane of (d2,d3). Updates m_fnh_h in place,
// produces new q1 (in place) and the q0 head-partial.
// ---------------------------------------------------------------------------
__global__ void __launch_bounds__(256) k_fnh(
    const float4* __restrict__ shstoh, const int* __restrict__ mh,
    const float* __restrict__ shoh,
    const float* __restrict__ q0in, const float* q1in, float* q1out,
    float* mhh, float* __restrict__ q0h, int first, int do_lsm) {
  __shared__ float4 s_x[4][256];           // 16 KB factor ring
  __shared__ float2 s_m[4][256];           //  8 KB old-message ring
  __shared__ int    s_k[4][256];           //  4 KB mask ring
  __shared__ float s_q1[T96 * 2], s_q0[T96 * 2];
  __shared__ float a_q0[8][T96 * 2], a_q1[8][T96 * 2];

  const int tid = threadIdx.x;
  const int w = tid >> 5, lane = tid & 31;
  const int bd = blockIdx.x;  // b*T + d1

  for (int i = tid; i < T96 * 2; i += 256) {
    s_q1[i] = q1in[(size_t)bd * T96 * 2 + i];
    s_q0[i] = q0in[(size_t)bd * T96 * 2 + i];
  }
  for (int i = tid; i < 8 * T96 * 2; i += 256) {
    (&a_q0[0][0])[i] = 0.0f;
    (&a_q1[0][0])[i] = 0.0f;
  }

  const char* xbase = (const char*)shstoh + (size_t)bd * NP * 16;
  char* mbase = (char*)mhh + (size_t)bd * NP * 8;
  const char* kbase = (const char*)(mh + (size_t)bd * NP);

  const unsigned lx = lds_off(&s_x[0][0]);
  const unsigned lm = lds_off(&s_m[0][0]);
  const unsigned lk = lds_off(&s_k[0][0]);

  // prologue: async-stage steps 0..2 (each wave consumes only its own lanes)
  for (int s = 0; s < 3; ++s) {
    const unsigned pp = (unsigned)(s * 256 + tid);
    async_b128(lx + (unsigned)s * 4096u + (unsigned)tid * 16u, pp * 16u, xbase);
    async_b64 (lm + (unsigned)s * 2048u + (unsigned)tid * 8u,  pp * 8u,  mbase);
    async_b32 (lk + (unsigned)s * 1024u + (unsigned)tid * 4u,  pp * 4u,  kbase);
  }
  __syncthreads();

  float* aq0 = a_q0[w];
  float* aq1 = a_q1[w];

  for (int i = 0; i < NSTEP; ++i) {
    const int buf = i & 3;
    if (i + 3 < NSTEP) {                 // keep 3 steps in flight
      const unsigned pn = (unsigned)((i + 3) * 256 + tid);
      const unsigned nb = (unsigned)((i + 3) & 3);
      async_b128(lx + nb * 4096u + (unsigned)tid * 16u, pn * 16u, xbase);
      async_b64 (lm + nb * 2048u + (unsigned)tid * 8u,  pn * 8u,  mbase);
      async_b32 (lk + nb * 1024u + (unsigned)tid * 4u,  pn * 4u,  kbase);
      wait_async_9();                    // step i retired (in-order loads)
    } else if (i + 3 == NSTEP) { wait_async_6(); }
    else if (i + 2 == NSTEP)   { wait_async_3(); }
    else                       { wait_async_0(); }

    const int p = i * 256 + tid;
    const int d2 = p / T96;
    const int d3 = p - d2 * T96;

    float f00, f01, f10, f11;
    edge_f(s_x[buf][tid], f00, f01, f10, f11);
    float2 mo = s_m[buf][tid];
    float mh0 = first ? -NLOG2 : mo.x;
    float mh1 = first ? -NLOG2 : mo.y;

    float mn0, mn1, h0, h1;
    bp_update(f00, f01, f10, f11,
              s_q1[d3 * 2 + 0] - mh0, s_q1[d3 * 2 + 1] - mh1,
              s_q0[d2 * 2 + 0], s_q0[d2 * 2 + 1],
              mn0, mn1, h0, h1);

    f2v hv; hv.x = h0; hv.y = h1;        // NT: keep L2 for factors/masks
    __builtin_nontemporal_store(hv, (f2v*)(mbase + (size_t)p * 8));

    float mk = (float)s_k[buf][tid];
    // q1 partial: sum over d2 -> per-wave slice, lanes hit distinct d3
    aq1[d3 * 2 + 0] += h0 * mk;
    aq1[d3 * 2 + 1] += h1 * mk;
    // q0 partial: sum over d3 -> deterministic wave shuffle-tree (lanes share d2)
    float s0 = mn0 * mk, s1 = mn1 * mk;
    for (int off = 16; off > 0; off >>= 1) {
      s0 += __shfl_xor(s0, off);
      s1 += __shfl_xor(s1, off);
    }
    if (lane == 0) { aq0[d2 * 2 + 0] += s0; aq0[d2 * 2 + 1] += s1; }
  }
  __syncthreads();

  if (tid < T96) {                      // q1[b,d1,d3] = s_shoh + sum
    const int d3 = tid;
    float acc0 = 0.0f, acc1 = 0.0f;
    for (int ww = 0; ww < 8; ++ww) { acc0 += a_q1[ww][d3 * 2 + 0]; acc1 += a_q1[ww][d3 * 2 + 1]; }
    float x0 = shoh[((size_t)bd * T96 + d3) * 2 + 0];
    float x1 = shoh[((size_t)bd * T96 + d3) * 2 + 1];
    float ls = lse2(x0, x1);
    float y0 = x0 - ls + acc0, y1 = x1 - ls + acc1;
    if (do_lsm) { float t = lse2(y0, y1); y0 -= t; y1 -= t; }
    q1out[((size_t)bd * T96 + d3) * 2 + 0] = y0;
    q1out[((size_t)bd * T96 + d3) * 2 + 1] = y1;
  } else if (tid < 2 * T96) {           // head partial for q0[b,d1,d2]
    const int d2 = tid - T96;
    float acc0 = 0.0f, acc1 = 0.0f;
    for (int ww = 0; ww < 8; ++ww) { acc0 += a_q0[ww][d2 * 2 + 0]; acc1 += a_q0[ww][d2 * 2 + 1]; }
    q0h[((size_t)bd * T96 + d2) * 2 + 0] = acc0;
    q0h[((size_t)bd * T96 + d2) * 2 + 1] = acc1;
  }
}

// ---------------------------------------------------------------------------
// fnt chain: one block per (b,d2) plane of (d1,d3). Updates m_fnt_t in place,
// produces new q2 (in place) and the q0 tail-partial.
// ---------------------------------------------------------------------------
__global__ void __launch_bounds__(256) k_fnt(
    const float4* __restrict__ shstot, const int* __restrict__ mt,
    const float* __restrict__ stot,
    const float* __restrict__ q0in, const float* q2in, float* q2out,
    float* mtt, float* __restrict__ q0t, int first, int do_lsm) {
  __shared__ float4 s_x[4][256];
  __shared__ float2 s_m[4][256];
  __shared__ int    s_k[4][256];
  __shared__ float s_q2[T96 * 2], s_q0[T96 * 2];
  __shared__ float a_q0[8][T96 * 2], a_q2[8][T96 * 2];

  const int tid = threadIdx.x;
  const int w = tid >> 5, lane = tid & 31;
  const int bd = blockIdx.x;  // b*T + d2
  const int b = bd / T96;
  const int d2 = bd - b * T96;

  for (int i = tid; i < T96 * 2; i += 256) s_q2[i] = q2in[(size_t)bd * T96 * 2 + i];
  if (tid < T96 * 2) {
    int d1 = tid >> 1, s = tid & 1;
    s_q0[tid] = q0in[(((size_t)(b * T96) + d1) * T96 + d2) * 2 + s];
  }
  for (int i = tid; i < 8 * T96 * 2; i += 256) {
    (&a_q0[0][0])[i] = 0.0f;
    (&a_q2[0][0])[i] = 0.0f;
  }

  const size_t ub = (size_t)b * T96 * T96 * T96 + (size_t)d2 * T96;  // pair index
  const char* xbase = (const char*)shstot + ub * 16;
  char* mbase = (char*)mtt + ub * 8;
  const char* kbase = (const char*)(mt + ub);

  const unsigned lx = lds_off(&s_x[0][0]);
  const unsigned lm = lds_off(&s_m[0][0]);
  const unsigned lk = lds_off(&s_k[0][0]);

  for (int s = 0; s < 3; ++s) {
    const int pp = s * 256 + tid;
    const int d1p = pp / T96, d3p = pp - d1p * T96;
    const unsigned vo = (unsigned)(d1p * NP + d3p);
    async_b128(lx + (unsigned)s * 4096u + (unsigned)tid * 16u, vo * 16u, xbase);
    async_b64 (lm + (unsigned)s * 2048u + (unsigned)tid * 8u,  vo * 8u,  mbase);
    async_b32 (lk + (unsigned)s * 1024u + (unsigned)tid * 4u,  vo * 4u,  kbase);
  }
  __syncthreads();

  float* aq0 = a_q0[w];
  float* aq2 = a_q2[w];

  for (int i = 0; i < NSTEP; ++i) {
    const int buf = i & 3;
    if (i + 3 < NSTEP) {
      const int pn = (i + 3) * 256 + tid;
      const int nd1 = pn / T96, nd3 = pn - nd1 * T96;
      const unsigned vo = (unsigned)(nd1 * NP + nd3);
      const unsigned nb = (unsigned)((i + 3) & 3);
      async_b128(lx + nb * 4096u + (unsigned)tid * 16u, vo * 16u, xbase);
      async_b64 (lm + nb * 2048u + (unsigned)tid * 8u,  vo * 8u,  mbase);
      async_b32 (lk + nb * 1024u + (unsigned)tid * 4u,  vo * 4u,  kbase);
      wait_async_9();
    } else if (i + 3 == NSTEP) { wait_async_6(); }
    else if (i + 2 == NSTEP)   { wait_async_3(); }
    else                       { wait_async_0(); }

    const int p = i * 256 + tid;
    const int d1 = p / T96;
    const int d3 = p - d1 * T96;
    const unsigned vo = (unsigned)(d1 * NP + d3);

    float f00, f01, f10, f11;
    edge_f(s_x[buf][tid], f00, f01, f10, f11);
    float2 mo = s_m[buf][tid];
    float mt0 = first ? -NLOG2 : mo.x;
    float mt1 = first ? -NLOG2 : mo.y;

    float mn0, mn1, h0, h1;
    bp_update(f00, f01, f10, f11,
              s_q2[d3 * 2 + 0] - mt0, s_q2[d3 * 2 + 1] - mt1,
              s_q0[d1 * 2 + 0], s_q0[d1 * 2 + 1],
              mn0, mn1, h0, h1);

    f2v hv; hv.x = h0; hv.y = h1;
    __builtin_nontemporal_store(hv, (f2v*)(mbase + (size_t)vo * 8));

    float mk = (float)s_k[buf][tid];
    aq2[d3 * 2 + 0] += h0 * mk;          // sum over d1
    aq2[d3 * 2 + 1] += h1 * mk;
    float s0 = mn0 * mk, s1 = mn1 * mk;  // sum over d3 (lanes share d1)
    for (int off = 16; off > 0; off >>= 1) {
      s0 += __shfl_xor(s0, off);
      s1 += __shfl_xor(s1, off);
    }
    if (lane == 0) { aq0[d1 * 2 + 0] += s0; aq0[d1 * 2 + 1] += s1; }
  }
  __syncthreads();

  if (tid < T96) {                      // q2[b,d2,d3] = s_stot + sum
    const int d3 = tid;
    float acc0 = 0.0f, acc1 = 0.0f;
    for (int ww = 0; ww < 8; ++ww) { acc0 += a_q2[ww][d3 * 2 + 0]; acc1 += a_q2[ww][d3 * 2 + 1]; }
    float x0 = stot[((size_t)bd * T96 + d3) * 2 + 0];
    float x1 = stot[((size_t)bd * T96 + d3) * 2 + 1];
    float ls = lse2(x0, x1);
    float y0 = x0 - ls + acc0, y1 = x1 - ls + acc1;
    if (do_lsm) { float t = lse2(y0, y1); y0 -= t; y1 -= t; }
    q2out[((size_t)bd * T96 + d3) * 2 + 0] = y0;
    q2out[((size_t)bd * T96 + d3) * 2 + 1] = y1;
  } else if (tid < 2 * T96) {           // tail partial for q0[b,d1,d2]
    const int d1 = tid - T96;
    float acc0 = 0.0f, acc1 = 0.0f;
    for (int ww = 0; ww < 8; ++ww) { acc0 += a_q0[ww][d1 * 2 + 0]; acc1 += a_q0[ww][d1 * 2 + 1]; }
    q0t[(((size_t)(b * T96) + d1) * T96 + d2) * 2 + 0] = acc0;
    q0t[(((size_t)(b * T96) + d1) * T96 + d2) * 2 + 1] = acc1;
  }
}

// q0 combine: q0 = log_softmax?(s_shst + head_partial + tail_partial)
__global__ void __launch_bounds__(256) k_q0c(
    const float* __restrict__ shst, const float* __restrict__ q0h,
    const float* __restrict__ q0t, float* __restrict__ q0out, int do_lsm) {
  int gid = blockIdx.x * 256 + threadIdx.x;
  if (gid >= B4 * T96 * T96) return;
  float x0 = shst[(size_t)gid * 2 + 0];
  float x1 = shst[(size_t)gid * 2 + 1];
  float ls = lse2(x0, x1);
  float y0 = x0 - ls + q0h[(size_t)gid * 2 + 0] + q0t[(size_t)gid * 2 + 0];
  float y1 = x1 - ls + q0h[(size_t)gid * 2 + 1] + q0t[(size_t)gid * 2 + 1];
  if (do_lsm) { float t = lse2(y0, y1); y0 -= t; y1 -= t; }
  q0out[(size_t)gid * 2 + 0] = y0;
  q0out[(size_t)gid * 2 + 1] = y1;
}

extern "C" void kernel_launch(void* const* d_in, const int* in_sizes, int n_in,
                              void* d_out, int out_size, void* d_ws, size_t ws_size,
                              hipStream_t stream) {
  (void)in_sizes; (void)n_in; (void)out_size; (void)ws_size;
  const float*  shst   = (const float*)d_in[0];
  const float*  shoh   = (const float*)d_in[1];
  const float*  stot   = (const float*)d_in[2];
  const float4* shstoh = (const float4*)d_in[3];
  const float4* shstot = (const float4*)d_in[4];
  // d_in[5]=mask (bool), d_in[6]=maskspan (bool): unused by the reference lbp
  const int* mh = (const int*)d_in[7];
  const int* mt = (const int*)d_in[8];

  float* out = (float*)d_out;
  float* out_q0 = out;
  float* out_q1 = out + NQ;
  float* out_q2 = out + 2 * NQ;

  const size_t nm = (size_t)B4 * T96 * T96 * T96 * 2;  // persistent message elems
  float* w = (float*)d_ws;
  float* mhh = w; w += nm;   // m_fnh_h  [B,T,T,T,2]
  float* mtt = w; w += nm;   // m_fnt_t  [B,T,T,T,2]
  float* q0  = w; w += NQ;
  float* q1  = w; w += NQ;
  float* q2  = w; w += NQ;
  float* q0h = w; w += NQ;   // head partial of q0
  float* q0t = w; w += NQ;   // tail partial of q0

  const int n3 = 3 * B4 * T96 * T96;
  k_init<<<(n3 + 255) / 256, 256, 0, stream>>>(mh, mt, q0, q1, q2);

  for (int it = 0; it < 3; ++it) {
    const int first = (it == 0);
    const int last = (it == 2);
    k_fnh<<<B4 * T96, 256, 0, stream>>>(shstoh, mh, shoh, q0, q1,
                                        last ? out_q1 : q1, mhh, q0h, first, !last);
    k_fnt<<<B4 * T96, 256, 0, stream>>>(shstot, mt, stot, q0, q2,
                                        last ? out_q2 : q2, mtt, q0t, first, !last);
    k_q0c<<<(B4 * T96 * T96 + 255) / 256, 256, 0, stream>>>(
        shst, q0h, q0t, last ? out_q0 : q0, !last);
  }
}